// Correlation_30760555774290
// MI455X (gfx1250) — compile-verified
//
#include <hip/hip_runtime.h>

typedef __attribute__((ext_vector_type(2))) float v2f;
typedef __attribute__((ext_vector_type(8))) float v8f;

#define C_CH   256
#define H_IMG  48
#define W_IMG  64
#define HW     3072            // H_IMG * W_IMG
#define GRID_D 21
#define NDISP  441

static __device__ __forceinline__ v8f wmma4(v2f a, v2f b, v8f c) {
    return __builtin_amdgcn_wmma_f32_16x16x4_f32(
        false, a, false, b, (short)0, c, false, false);
}

// Row-packed band GEMM.
//   For in2 row r, the chain of pairs (i, y = r + 20 - 2*i) all correlate
//   against row r.  One 16x16 M-tile packs 8 same-parity x-pixels of chain
//   position q (rows 0-7) and the SAME x-pixels of position q+1 (rows 8-15);
//   both halves share one B fragment B[c,n] = in2[b,c,r, x0 + 2n - 20] with
//   n = (m&7) + j in [0,28) -> only 2 N-tiles of 16 (vs 3 unpacked).
// One block  = (b, r, kx) handling chain positions 4kx .. 4kx+3 (2 M-tiles,
//   which share the B fragment -> 8 loads : 4 WMMAs per K-step).
// One wave   = (parity p, x-subgroup g of 8 same-parity pixels).
__global__ __launch_bounds__(256)
void corr_wmma_pack_kernel(const float* __restrict__ in1,
                           const float* __restrict__ in2,
                           float* __restrict__ out)
{
    const int kx = blockIdx.x;              // 0..5
    const int r  = (int)blockIdx.y - 20;    // in2 source row, -20..67
    const int b  = blockIdx.z;

    // chain bounds: i such that y = r + 20 - 2i lies in [0, 48)
    const int i_lo = (r > 27) ? ((r - 26) >> 1) : 0;
    const int i_hi = min(20, (r + 20) >> 1);
    const int L    = i_hi - i_lo + 1;       // 1..21

    const int q0 = 4 * kx;                  // first chain position here
    if (q0 >= L) return;                    // uniform over block, no barriers

    const bool g1v = (q0 + 1) < L;
    const bool g2v = (q0 + 2) < L;
    const bool g3v = (q0 + 3) < L;

    const int i0 = i_lo + q0;
    const int y0 = r + 20 - 2 * i0;         // always valid when q0 < L
    const int y1 = g1v ? (y0 - 2) : y0;     // invalid groups alias y0;
    const int y2 = g2v ? (y0 - 4) : y0;     // their results are never stored
    const int y3 = g3v ? (y0 - 6) : y0;

    const int wave = threadIdx.x >> 5;      // 0..7
    const int lane = threadIdx.x & 31;
    const int p = wave & 1;                 // x parity
    const int g = wave >> 1;                // subgroup of 8 same-parity pixels

    const bool r_ok = (r >= 0) && (r < H_IMG);
    const int  rs   = r_ok ? r : 0;

    const int half = lane >> 4;             // K half (A/B) / M half (D)
    const int l16  = lane & 15;
    const int kb   = 2 * half;              // channel sub-base within K=4 block

    // ---- A fragments (ISA layout: lane holds row m = lane&15, K = kb+vgpr) --
    const int mrow = l16 & 7;               // row within its (y,i) group
    const int x1   = p + 2 * (8 * g + mrow);         // in [0, 64) always
    const int yA0  = (l16 < 8) ? y0 : y1;            // tile0: groups q0,q0+1
    const int yA1  = (l16 < 8) ? y2 : y3;            // tile1: groups q0+2,q0+3
    const float* pA0 = in1 + ((((size_t)b * C_CH + kb) * H_IMG + yA0) * W_IMG + x1);
    const float* pA1 = in1 + ((((size_t)b * C_CH + kb) * H_IMG + yA1) * W_IMG + x1);

    // ---- B fragment, shared by both tiles: col n = l16 (+16 for s=1) --------
    const size_t rowbase = (((size_t)b * C_CH + kb) * H_IMG + rs) * W_IMG;
    const int u0 = p + 2 * (8 * g + l16) - 20;
    const int u1 = u0 + 32;
    const bool ok0 = (u0 >= 0) && (u0 < W_IMG);
    const bool ok1 = (u1 >= 0) && (u1 < W_IMG);
    const float* pB0 = in2 + rowbase + (ok0 ? u0 : 0);
    const float* pB1 = in2 + rowbase + (ok1 ? u1 : 0);
    const float mk0 = ok0 ? 1.0f : 0.0f;    // branch-free edge masking keeps
    const float mk1 = ok1 ? 1.0f : 0.0f;    // EXEC all-ones around WMMA

    v8f acc00 = {}, acc01 = {}, acc10 = {}, acc11 = {};

    if (r_ok) {
        if (g2v) {  // both M-tiles live: 4 WMMAs share 2 A + 2 B fragments
            #pragma unroll 4
            for (int kk = 0; kk < C_CH / 4; ++kk) {
                v2f a0, a1, b0, b1;
                a0.x = pA0[0];        a0.y = pA0[HW];
                a1.x = pA1[0];        a1.y = pA1[HW];
                b0.x = pB0[0] * mk0;  b0.y = pB0[HW] * mk0;
                b1.x = pB1[0] * mk1;  b1.y = pB1[HW] * mk1;
                acc00 = wmma4(a0, b0, acc00);
                acc01 = wmma4(a0, b1, acc01);
                acc10 = wmma4(a1, b0, acc10);
                acc11 = wmma4(a1, b1, acc11);
                pA0 += 4 * HW; pA1 += 4 * HW;
                pB0 += 4 * HW; pB1 += 4 * HW;
            }
        } else {    // only tile0
            #pragma unroll 4
            for (int kk = 0; kk < C_CH / 4; ++kk) {
                v2f a0, b0, b1;
                a0.x = pA0[0];        a0.y = pA0[HW];
                b0.x = pB0[0] * mk0;  b0.y = pB0[HW] * mk0;
                b1.x = pB1[0] * mk1;  b1.y = pB1[HW] * mk1;
                acc00 = wmma4(a0, b0, acc00);
                acc01 = wmma4(a0, b1, acc01);
                pA0 += 4 * HW;
                pB0 += 4 * HW; pB1 += 4 * HW;
            }
        }
    }

    // ---- band extraction + store --------------------------------------------
    // D layout: lane, vgpr v -> M = v + 8*half, N = l16.  Row-in-group = v,
    // j = n - (m&7) = 16*s + l16 - v, displacement d = i*21 + j.
    const float scale = 1.0f / (float)C_CH;
    const int   xbase = p + 16 * g;          // x = xbase + 2*v

    {   // tile 0: half 0 -> (i0, y0), half 1 -> (i0+1, y1)
        const int  i_o = half ? (i0 + 1) : i0;
        const int  y_o = half ? y1 : y0;
        const bool hv  = half ? g1v : true;
        if (hv) {
            #pragma unroll
            for (int v = 0; v < 8; ++v) {
                const int x = xbase + 2 * v;
                int j = l16 - v;                                   // s = 0
                if (j >= 0 && j < GRID_D)
                    out[(((size_t)b * NDISP + i_o * GRID_D + j) * H_IMG + y_o) * W_IMG + x]
                        = acc00[v] * scale;
                j = 16 + l16 - v;                                  // s = 1 (j>=9)
                if (j < GRID_D)
                    out[(((size_t)b * NDISP + i_o * GRID_D + j) * H_IMG + y_o) * W_IMG + x]
                        = acc01[v] * scale;
            }
        }
    }
    if (g2v) {  // tile 1: half 0 -> (i0+2, y2), half 1 -> (i0+3, y3)
        const int  i_o = half ? (i0 + 3) : (i0 + 2);
        const int  y_o = half ? y3 : y2;
        const bool hv  = half ? g3v : true;
        if (hv) {
            #pragma unroll
            for (int v = 0; v < 8; ++v) {
                const int x = xbase + 2 * v;
                int j = l16 - v;                                   // s = 0
                if (j >= 0 && j < GRID_D)
                    out[(((size_t)b * NDISP + i_o * GRID_D + j) * H_IMG + y_o) * W_IMG + x]
                        = acc10[v] * scale;
                j = 16 + l16 - v;                                  // s = 1
                if (j < GRID_D)
                    out[(((size_t)b * NDISP + i_o * GRID_D + j) * H_IMG + y_o) * W_IMG + x]
                        = acc11[v] * scale;
            }
        }
    }
}

extern "C" void kernel_launch(void* const* d_in, const int* in_sizes, int n_in,
                              void* d_out, int out_size, void* d_ws, size_t ws_size,
                              hipStream_t stream)
{
    const float* in1 = (const float*)d_in[0];
    const float* in2 = (const float*)d_in[1];
    float* out = (float*)d_out;

    // blocks: (chain tile-pair kx, in2 row r+20, batch b); 256 threads = 8 waves
    dim3 grid(6, 88, 16);
    corr_wmma_pack_kernel<<<grid, 256, 0, stream>>>(in1, in2, out);
}